// GCNLinkPredictor_77481210020188
// MI455X (gfx1250) — compile-verified
//
#include <hip/hip_runtime.h>
#include <hip/hip_bf16.h>
#include <math.h>

#define N_NODES 100000
#define N_EDGES 3200000
#define F 128
#define N_TILES (N_NODES / 16)   // 6250 exactly

typedef __attribute__((ext_vector_type(2))) float v2f;
typedef __attribute__((ext_vector_type(8))) float v8f;

// ---------------------------------------------------------------------------
// 1) deg[i] = 1  (self-loop contribution pre-counted)
__global__ __launch_bounds__(256) void k_init_deg(unsigned* __restrict__ deg) {
    int i = blockIdx.x * blockDim.x + threadIdx.x;
    if (i < N_NODES) deg[i] = 1u;
}

// 2) deg[dst[e]] += 1 for real edges
__global__ __launch_bounds__(256) void k_count(const int* __restrict__ dst,
                                               unsigned* __restrict__ deg) {
    int e = blockIdx.x * blockDim.x + threadIdx.x;
    if (e < N_EDGES) atomicAdd(&deg[dst[e]], 1u);
}

// 3) dinv[i] = rsqrt(deg[i])   (deg >= 1 always, no zero branch needed)
__global__ __launch_bounds__(256) void k_dinv(const unsigned* __restrict__ deg,
                                              float* __restrict__ dinv) {
    int i = blockIdx.x * blockDim.x + threadIdx.x;
    if (i < N_NODES) dinv[i] = rsqrtf((float)deg[i]);
}

// ---------------------------------------------------------------------------
// 4) h = x @ W  via V_WMMA_F32_16X16X4_F32.
//    One wave32 per 16-row tile; 8 column tiles (all 128 outputs) per wave.
//    A 16x4 f32 frag: lanes 0-15 = rows, VGPR j = K=j; lanes 16-31 = K=2+j.
//    B 4x16 f32 frag: lanes 0-15 = cols, VGPR j = K=j; lanes 16-31 = K=2+j.
//    C/D 16x16 f32: VGPR r, lanes 0-15 -> M=r, lanes 16-31 -> M=8+r.
__global__ __launch_bounds__(128) void k_gemm(const float* __restrict__ x,
                                              const float* __restrict__ W,
                                              float* __restrict__ h) {
    const int wave = threadIdx.x >> 5;
    const int lane = threadIdx.x & 31;
    const int tile = blockIdx.x * 4 + wave;      // wave-uniform
    if (tile >= N_TILES) return;                 // whole wave exits together

    const int lrow  = lane & 15;
    const int khalf = (lane >> 4) * 2;           // 0 or 2
    const float* xrow = x + (size_t)(tile * 16 + lrow) * F;

    v8f acc[8] = {};                             // zero-init accumulators

    for (int k0 = 0; k0 < F; k0 += 4) {
        const float2 av = *(const float2*)(xrow + k0 + khalf);  // 8B aligned
        v2f a; a.x = av.x; a.y = av.y;
#pragma unroll
        for (int nt = 0; nt < 8; ++nt) {
            const int col = nt * 16 + lrow;
            v2f bf;
            bf.x = W[(size_t)(k0 + khalf + 0) * F + col];
            bf.y = W[(size_t)(k0 + khalf + 1) * F + col];
            acc[nt] = __builtin_amdgcn_wmma_f32_16x16x4_f32(
                false, a, false, bf, (short)0, acc[nt], false, false);
        }
    }

    const int mbase = tile * 16 + ((lane >> 4) ? 8 : 0);
#pragma unroll
    for (int nt = 0; nt < 8; ++nt) {
        const int col = nt * 16 + lrow;
#pragma unroll
        for (int r = 0; r < 8; ++r)
            h[(size_t)(mbase + r) * F + col] = acc[nt][r];
    }
}

// ---------------------------------------------------------------------------
// 5) agg[n] = b + h[n] * dinv[n]^2   (bias + self-loop message, no atomics)
//    One wave per node, float4 per lane.
__global__ __launch_bounds__(256) void k_selfbias(const float* __restrict__ h,
                                                  const float* __restrict__ dinv,
                                                  const float* __restrict__ b,
                                                  float* __restrict__ agg) {
    const int wid  = blockIdx.x * (blockDim.x >> 5) + (threadIdx.x >> 5);
    const int lane = threadIdx.x & 31;
    if (wid >= N_NODES) return;
    const float dn = dinv[wid];
    const float s  = dn * dn;
    const float4 hv = *(const float4*)(h + (size_t)wid * F + lane * 4);
    const float4 bv = *(const float4*)(b + lane * 4);
    float4 o;
    o.x = bv.x + hv.x * s;  o.y = bv.y + hv.y * s;
    o.z = bv.z + hv.z * s;  o.w = bv.w + hv.w * s;
    *(float4*)(agg + (size_t)wid * F + lane * 4) = o;
}

// ---------------------------------------------------------------------------
// 6) agg[dst] += h[src] * dinv[src]*dinv[dst]
//    One wave per edge: coalesced 512B row gather + native f32 atomic scatter.
//    h and agg both fit in the 192MB L2 -> atomics resolve in L2.
__global__ __launch_bounds__(256) void k_scatter(const int* __restrict__ src,
                                                 const int* __restrict__ dst,
                                                 const float* __restrict__ dinv,
                                                 const float* __restrict__ h,
                                                 float* __restrict__ agg) {
    const int wid  = blockIdx.x * (blockDim.x >> 5) + (threadIdx.x >> 5);
    const int lane = threadIdx.x & 31;
    if (wid >= N_EDGES) return;
    const int   s    = src[wid];
    const int   d    = dst[wid];
    const float norm = dinv[s] * dinv[d];
    const float4 hv = *(const float4*)(h + (size_t)s * F + lane * 4);
    float* out = agg + (size_t)d * F + lane * 4;
    unsafeAtomicAdd(out + 0, hv.x * norm);
    unsafeAtomicAdd(out + 1, hv.y * norm);
    unsafeAtomicAdd(out + 2, hv.z * norm);
    unsafeAtomicAdd(out + 3, hv.w * norm);
}

// ---------------------------------------------------------------------------
// 7) out[n] = sigmoid( relu(agg[n]) . fc_w + fc_b )  -- wave32 shuffle reduce
__global__ __launch_bounds__(256) void k_final(const float* __restrict__ agg,
                                               const float* __restrict__ fcw,
                                               const float* __restrict__ fcb,
                                               float* __restrict__ out) {
    const int wid  = blockIdx.x * (blockDim.x >> 5) + (threadIdx.x >> 5);
    const int lane = threadIdx.x & 31;
    if (wid >= N_NODES) return;
    const float4 av = *(const float4*)(agg + (size_t)wid * F + lane * 4);
    const float4 wv = *(const float4*)(fcw + lane * 4);
    float sum = fmaxf(av.x, 0.f) * wv.x + fmaxf(av.y, 0.f) * wv.y +
                fmaxf(av.z, 0.f) * wv.z + fmaxf(av.w, 0.f) * wv.w;
#pragma unroll
    for (int off = 16; off > 0; off >>= 1)
        sum += __shfl_down(sum, off, 32);
    if (lane == 0) {
        const float z = sum + fcb[0];
        out[wid] = 1.0f / (1.0f + __expf(-z));
    }
}

// ---------------------------------------------------------------------------
static inline size_t rnd512(size_t x) { return (x + 511) & ~(size_t)511; }

extern "C" void kernel_launch(void* const* d_in, const int* in_sizes, int n_in,
                              void* d_out, int out_size, void* d_ws, size_t ws_size,
                              hipStream_t stream) {
    const float* x   = (const float*)d_in[0];
    const int*   ei  = (const int*)  d_in[1];
    const float* W   = (const float*)d_in[2];
    const float* b   = (const float*)d_in[3];
    const float* fcw = (const float*)d_in[4];
    const float* fcb = (const float*)d_in[5];
    float* out = (float*)d_out;

    char* ws = (char*)d_ws;
    size_t off = 0;
    unsigned* deg  = (unsigned*)(ws + off); off += rnd512((size_t)N_NODES * 4);
    float*    dinv = (float*)   (ws + off); off += rnd512((size_t)N_NODES * 4);
    float*    h    = (float*)   (ws + off); off += rnd512((size_t)N_NODES * F * 4);
    float*    agg  = (float*)   (ws + off);

    const int* src = ei;            // edge_index[0]
    const int* dst = ei + N_EDGES;  // edge_index[1]

    k_init_deg<<<(N_NODES + 255) / 256, 256, 0, stream>>>(deg);
    k_count   <<<(N_EDGES + 255) / 256, 256, 0, stream>>>(dst, deg);
    k_dinv    <<<(N_NODES + 255) / 256, 256, 0, stream>>>(deg, dinv);
    k_gemm    <<<(N_TILES + 3) / 4, 128, 0, stream>>>(x, W, h);
    k_selfbias<<<(N_NODES + 7) / 8, 256, 0, stream>>>(h, dinv, b, agg);
    k_scatter <<<(N_EDGES + 7) / 8, 256, 0, stream>>>(src, dst, dinv, h, agg);
    k_final   <<<(N_NODES + 7) / 8, 256, 0, stream>>>(agg, fcw, fcb, out);
}